// SudokuGCN_64862596104804
// MI455X (gfx1250) — compile-verified
//
#include <hip/hip_runtime.h>
#include <hip/hip_bf16.h>
#include <math.h>

typedef __attribute__((ext_vector_type(2))) float v2f;
typedef __attribute__((ext_vector_type(8))) float v8f;

// ---------------------------------------------------------------------------
// Utility kernels
// ---------------------------------------------------------------------------
__global__ void fill_f32(float* __restrict__ p, float v, long long n) {
    long long i = (long long)blockIdx.x * blockDim.x + threadIdx.x;
    if (i < n) p[i] = v;
}

__global__ void degree_kernel(const int* __restrict__ col, float* __restrict__ deg,
                              long long E) {
    long long e = (long long)blockIdx.x * blockDim.x + threadIdx.x;
    if (e < E) unsafeAtomicAdd(deg + col[e], 1.0f);
}

__global__ void rsqrt_inplace(float* __restrict__ d, long long n) {
    long long i = (long long)blockIdx.x * blockDim.x + threadIdx.x;
    if (i < n) d[i] = rsqrtf(d[i]);   // deg >= 1 always (self-loops)
}

// ---------------------------------------------------------------------------
// Dense GEMM H[N x FOUT] = X[N x K] @ W[K x FOUT] using V_WMMA_F32_16X16X4_F32
// One wave per 16x16 output tile. f32 in / f32 accumulate == reference math.
// K/FOUT tails handled by clamped (always in-bounds) loads + lane selects, so
// no divergent-branch predication appears in front of the WMMAs.
// ---------------------------------------------------------------------------
template <int K, int FOUT>
__global__ void gemm_wmma_f32(const float* __restrict__ X,
                              const float* __restrict__ W,
                              float* __restrict__ H, int nrows) {
    constexpr int NT = (FOUT + 15) / 16;           // n-tiles
    const int lane = threadIdx.x & 31;
    const int wave = threadIdx.x >> 5;
    const int tile = blockIdx.x * (blockDim.x >> 5) + wave;
    const int mt = tile / NT;
    const int nt = tile % NT;
    const int m0 = mt * 16;
    const int n0 = nt * 16;
    if (m0 >= nrows) return;                       // wave-uniform exit

    const int half = lane >> 4;                    // 0 or 1
    const int lid  = lane & 15;

    const int m  = m0 + lid;
    const int n  = n0 + lid;
    const int nc = (n < FOUT) ? n : (FOUT - 1);    // clamped column (in-bounds)
    const bool nok = (n < FOUT);

    v8f acc = {};
#pragma unroll
    for (int k0 = 0; k0 < K; k0 += 4) {
        const int kb  = k0 + half * 2;             // this lane's K pair
        const int kb0 = (kb     < K) ? kb     : (K - 1);
        const int kb1 = (kb + 1 < K) ? kb + 1 : (K - 1);
        v2f a, b;
        const float ax = X[(long long)m * K + kb0];
        const float ay = X[(long long)m * K + kb1];
        const float bx = W[kb0 * FOUT + nc];
        const float by = W[kb1 * FOUT + nc];
        a.x = (kb     < K) ? ax : 0.0f;
        a.y = (kb + 1 < K) ? ay : 0.0f;
        b.x = (nok && kb     < K) ? bx : 0.0f;
        b.y = (nok && kb + 1 < K) ? by : 0.0f;
        acc = __builtin_amdgcn_wmma_f32_16x16x4_f32(
            /*neg_a=*/false, a, /*neg_b=*/false, b,
            /*c_mod=*/(short)0, acc, /*reuse_a=*/false, /*reuse_b=*/false);
    }

    if (nok) {
#pragma unroll
        for (int r = 0; r < 8; ++r) {
            const int mm = m0 + r + half * 8;      // C/D layout, ISA 7.12.2
            H[(long long)mm * FOUT + n] = acc[r];
        }
    }
}

// ---------------------------------------------------------------------------
// Edge-parallel scatter: out[col] += h[row] * dis[row]*dis[col]
// LPE lanes per edge, lane == feature -> coalesced gather + native f32 atomics
// ---------------------------------------------------------------------------
template <int F, int LPE>
__global__ void gcn_scatter(const int* __restrict__ row, const int* __restrict__ col,
                            const float* __restrict__ dis, const float* __restrict__ h,
                            float* __restrict__ out, long long E) {
    const long long idx = (long long)blockIdx.x * blockDim.x + threadIdx.x;
    const long long e  = idx / LPE;
    const int       f0 = (int)(idx % LPE);
    if (e >= E) return;
    const int r = row[e];
    const int c = col[e];
    const float norm = dis[r] * dis[c];
#pragma unroll
    for (int f = f0; f < F; f += LPE) {
        unsafeAtomicAdd(out + (long long)c * F + f,
                        h[(long long)r * F + f] * norm);
    }
}

// F=64 specialization: one wave per edge, float2 gather (global_load_b64),
// two native f32 atomics per lane.
__global__ void gcn_scatter64(const int* __restrict__ row, const int* __restrict__ col,
                              const float* __restrict__ dis, const float* __restrict__ h,
                              float* __restrict__ out, long long E) {
    const long long idx = (long long)blockIdx.x * blockDim.x + threadIdx.x;
    const long long e  = idx >> 5;
    const int     lane = (int)(idx & 31);
    if (e >= E) return;
    const int r = row[e];
    const int c = col[e];
    const float norm = dis[r] * dis[c];
    const float2 hv = *(const float2*)(h + (long long)r * 64 + lane * 2);
    float* dst = out + (long long)c * 64 + lane * 2;
    unsafeAtomicAdd(dst,     hv.x * norm);
    unsafeAtomicAdd(dst + 1, hv.y * norm);
}

// ---------------------------------------------------------------------------
// Fused epilogue: agg += self-loop term + bias, optional ReLU (in place)
// ---------------------------------------------------------------------------
template <int F, bool RELU>
__global__ void epilogue_kernel(float* __restrict__ agg, const float* __restrict__ h,
                                const float* __restrict__ dis, const float* __restrict__ b,
                                long long n) {
    const long long i = (long long)blockIdx.x * blockDim.x + threadIdx.x;
    if (i >= n * F) return;
    const long long node = i / F;
    const int f = (int)(i % F);
    const float d = dis[node];
    float v = agg[i] + h[i] * d * d + b[f];
    if (RELU) v = fmaxf(v, 0.0f);
    agg[i] = v;
}

// ---------------------------------------------------------------------------
// Final layer: self-loop + bias + log_softmax over F=10 features per node
// ---------------------------------------------------------------------------
__global__ void final_logsoftmax(const float* __restrict__ agg, const float* __restrict__ h,
                                 const float* __restrict__ dis, const float* __restrict__ b,
                                 float* __restrict__ out, long long n) {
    const long long node = (long long)blockIdx.x * blockDim.x + threadIdx.x;
    if (node >= n) return;
    const float d  = dis[node];
    const float dd = d * d;
    float v[10];
    float mx = -INFINITY;
#pragma unroll
    for (int f = 0; f < 10; ++f) {
        v[f] = agg[node * 10 + f] + h[node * 10 + f] * dd + b[f];
        mx = fmaxf(mx, v[f]);
    }
    float s = 0.0f;
#pragma unroll
    for (int f = 0; f < 10; ++f) s += expf(v[f] - mx);
    const float lse = mx + logf(s);
#pragma unroll
    for (int f = 0; f < 10; ++f) out[node * 10 + f] = v[f] - lse;
}

// ---------------------------------------------------------------------------
// Host orchestration
// ---------------------------------------------------------------------------
static inline int cdiv_ll(long long a, long long b) { return (int)((a + b - 1) / b); }

extern "C" void kernel_launch(void* const* d_in, const int* in_sizes, int n_in,
                              void* d_out, int out_size, void* d_ws, size_t ws_size,
                              hipStream_t stream) {
    const float* x   = (const float*)d_in[0];
    const int*   ei  = (const int*)d_in[1];
    const float* W1  = (const float*)d_in[2];
    const float* b1  = (const float*)d_in[3];
    const float* W2  = (const float*)d_in[4];
    const float* b2  = (const float*)d_in[5];
    const float* W3  = (const float*)d_in[6];
    const float* b3  = (const float*)d_in[7];
    float*       out = (float*)d_out;

    const long long N = in_sizes[0] / 10;   // 162000 (divisible by 16)
    const long long E = in_sizes[1] / 2;    // 3240000
    const int* row = ei;                    // edge_index[0] = source
    const int* col = ei + E;                // edge_index[1] = target

    // Workspace layout (floats): dis[N] | hbuf[N*64] | abuf[N*64]  (~83.6 MB)
    float* ws   = (float*)d_ws;
    float* dis  = ws;
    float* hbuf = ws + N;
    float* abuf = hbuf + N * 64;

    const int B = 256;

    // --- degrees & normalization (deg starts at 1.0 = self-loop) ---
    fill_f32<<<cdiv_ll(N, B), B, 0, stream>>>(dis, 1.0f, N);
    degree_kernel<<<cdiv_ll(E, B), B, 0, stream>>>(col, dis, E);
    rsqrt_inplace<<<cdiv_ll(N, B), B, 0, stream>>>(dis, N);

    const int mt = (int)(N / 16);           // 10125 m-tiles
    const int WPB = 4;                      // waves per block (blockDim 128)

    // ================= Layer 1: 10 -> 32 =================
    gemm_wmma_f32<10, 32><<<cdiv_ll((long long)mt * 2, WPB), WPB * 32, 0, stream>>>(
        x, W1, hbuf, (int)N);
    fill_f32<<<cdiv_ll(N * 32, B), B, 0, stream>>>(abuf, 0.0f, N * 32);
    gcn_scatter<32, 32><<<cdiv_ll(E * 32, B), B, 0, stream>>>(row, col, dis, hbuf, abuf, E);
    epilogue_kernel<32, true><<<cdiv_ll(N * 32, B), B, 0, stream>>>(abuf, hbuf, dis, b1, N);

    // ================= Layer 2: 32 -> 64 =================
    gemm_wmma_f32<32, 64><<<cdiv_ll((long long)mt * 4, WPB), WPB * 32, 0, stream>>>(
        abuf, W2, hbuf, (int)N);
    fill_f32<<<cdiv_ll(N * 64, B), B, 0, stream>>>(abuf, 0.0f, N * 64);
    gcn_scatter64<<<cdiv_ll(E * 32, B), B, 0, stream>>>(row, col, dis, hbuf, abuf, E);
    epilogue_kernel<64, true><<<cdiv_ll(N * 64, B), B, 0, stream>>>(abuf, hbuf, dis, b2, N);

    // ================= Layer 3: 64 -> 10 =================
    gemm_wmma_f32<64, 10><<<cdiv_ll((long long)mt * 1, WPB), WPB * 32, 0, stream>>>(
        abuf, W3, hbuf, (int)N);
    fill_f32<<<cdiv_ll(N * 10, B), B, 0, stream>>>(abuf, 0.0f, N * 10);
    gcn_scatter<10, 16><<<cdiv_ll(E * 16, B), B, 0, stream>>>(row, col, dis, hbuf, abuf, E);
    final_logsoftmax<<<cdiv_ll(N, B), B, 0, stream>>>(abuf, hbuf, dis, b3, out, N);
}